// FastAttention_16561393893724
// MI455X (gfx1250) — compile-verified
//
#include <hip/hip_runtime.h>
#include <cmath>

typedef float v2f __attribute__((ext_vector_type(2)));
typedef float v8f __attribute__((ext_vector_type(8)));

#define BATCH   4
#define NSEQ    8192
#define DIM_C   512
#define HEADS_C 8
#define DHEAD   64
#define QKV_W   1536
#define BH      (BATCH * HEADS_C)   // 32
#define SCALE   0.125f              // 64^-0.5

// ------------------------------------------------------------------
// GEMM: C[M,N] = A[M,K] @ B[K,N] (+ bias[N]), fp32 via V_WMMA_F32_16X16X4_F32
// Block tile 128x128, K-tile 32, double-buffered LDS (one barrier / K-block).
// 256 threads = 8 waves (wave32), waves 4(M)x2(N); each wave owns 32x64 =
// 2x4 WMMA 16x16 tiles (8 wmma per k-step vs 6 LDS b64 fragment loads).
// ------------------------------------------------------------------
#define BM 128
#define BN 128
#define BK 32
#define LDA_S 36   // A tile LDS stride (floats): 144B rows -> float4-aligned
#define LDB_S 34   // Bt tile LDS stride (floats): even -> 8B-aligned b64 reads

__global__ __launch_bounds__(256) void gemm_wmma_f32(
    const float* __restrict__ A, const float* __restrict__ B,
    const float* __restrict__ bias, float* __restrict__ C,
    int M, int N, int K)
{
  __shared__ __align__(16) float As[2][BM * LDA_S];
  __shared__ __align__(16) float Bs[2][BN * LDB_S];   // transposed: Bs[n][k]

  const int tid  = threadIdx.x;
  const int wave = tid >> 5;
  const int lane = tid & 31;
  const int hf   = lane >> 4;    // half of the wave -> K group (A/B), M group (D)
  const int l16  = lane & 15;    // row (A:M, B:N, D:N)

  const int blockM = blockIdx.y * BM;
  const int blockN = blockIdx.x * BN;
  const int wm = (wave >> 1) * 32;   // wave M offset: 0/32/64/96
  const int wn = (wave & 1)  * 64;   // wave N offset: 0/64

  // A-tile staging indices: 128x32 = 1024 float4, 4/thread
  const int arow[4] = { (tid + 0*256) >> 3, (tid + 1*256) >> 3,
                        (tid + 2*256) >> 3, (tid + 3*256) >> 3 };
  const int ac4 = tid & 7;
  // B-tile staging indices: 32x128 = 1024 float4, 4/thread
  const int brow[4] = { (tid + 0*256) >> 5, (tid + 1*256) >> 5,
                        (tid + 2*256) >> 5, (tid + 3*256) >> 5 };
  const int bc4 = tid & 31;

  const v8f vzero = {0.f,0.f,0.f,0.f,0.f,0.f,0.f,0.f};
  v8f acc[2][4];
#pragma unroll
  for (int i = 0; i < 2; ++i)
#pragma unroll
    for (int j = 0; j < 4; ++j) acc[i][j] = vzero;

  // ---- prologue: stage tile for k0 = 0 into buffer 0 ----
#pragma unroll
  for (int i = 0; i < 4; ++i) {
    float4 t = *(const float4*)&A[(size_t)(blockM + arow[i]) * K + ac4 * 4];
    *(float4*)&As[0][arow[i] * LDA_S + ac4 * 4] = t;
  }
#pragma unroll
  for (int i = 0; i < 4; ++i) {
    float4 t = *(const float4*)&B[(size_t)brow[i] * N + blockN + bc4 * 4];
    Bs[0][(bc4 * 4 + 0) * LDB_S + brow[i]] = t.x;
    Bs[0][(bc4 * 4 + 1) * LDB_S + brow[i]] = t.y;
    Bs[0][(bc4 * 4 + 2) * LDB_S + brow[i]] = t.z;
    Bs[0][(bc4 * 4 + 3) * LDB_S + brow[i]] = t.w;
  }
  __syncthreads();

  int cur = 0;
  for (int k0 = 0; k0 < K; k0 += BK) {
    const int nxt = cur ^ 1;
    const bool more = (k0 + BK) < K;

    // ---- issue global loads for the NEXT tile (overlap with compute) ----
    float4 ta[4], tb[4];
    if (more) {
#pragma unroll
      for (int i = 0; i < 4; ++i)
        ta[i] = *(const float4*)&A[(size_t)(blockM + arow[i]) * K + (k0 + BK) + ac4 * 4];
#pragma unroll
      for (int i = 0; i < 4; ++i)
        tb[i] = *(const float4*)&B[(size_t)(k0 + BK + brow[i]) * N + blockN + bc4 * 4];
    }

    // ---- compute on current buffer ----
    // ISA layout 16x4 f32 A / 4x16 B: lane holds M(N)=l16, K = 2*hf + e
#pragma unroll
    for (int kk = 0; kk < BK; kk += 4) {
      const int ka = kk + 2 * hf;
      v2f a0 = *(const v2f*)&As[cur][(wm +      l16) * LDA_S + ka];
      v2f a1 = *(const v2f*)&As[cur][(wm + 16 + l16) * LDA_S + ka];
      v2f b0 = *(const v2f*)&Bs[cur][(wn +      l16) * LDB_S + ka];
      v2f b1 = *(const v2f*)&Bs[cur][(wn + 16 + l16) * LDB_S + ka];
      v2f b2 = *(const v2f*)&Bs[cur][(wn + 32 + l16) * LDB_S + ka];
      v2f b3 = *(const v2f*)&Bs[cur][(wn + 48 + l16) * LDB_S + ka];
      acc[0][0] = __builtin_amdgcn_wmma_f32_16x16x4_f32(false, a0, false, b0,
                    (short)0, acc[0][0], false, false);
      acc[0][1] = __builtin_amdgcn_wmma_f32_16x16x4_f32(false, a0, false, b1,
                    (short)0, acc[0][1], false, false);
      acc[0][2] = __builtin_amdgcn_wmma_f32_16x16x4_f32(false, a0, false, b2,
                    (short)0, acc[0][2], false, false);
      acc[0][3] = __builtin_amdgcn_wmma_f32_16x16x4_f32(false, a0, false, b3,
                    (short)0, acc[0][3], false, false);
      acc[1][0] = __builtin_amdgcn_wmma_f32_16x16x4_f32(false, a1, false, b0,
                    (short)0, acc[1][0], false, false);
      acc[1][1] = __builtin_amdgcn_wmma_f32_16x16x4_f32(false, a1, false, b1,
                    (short)0, acc[1][1], false, false);
      acc[1][2] = __builtin_amdgcn_wmma_f32_16x16x4_f32(false, a1, false, b2,
                    (short)0, acc[1][2], false, false);
      acc[1][3] = __builtin_amdgcn_wmma_f32_16x16x4_f32(false, a1, false, b3,
                    (short)0, acc[1][3], false, false);
    }

    // ---- store staged registers into the other buffer, single barrier ----
    if (more) {
#pragma unroll
      for (int i = 0; i < 4; ++i)
        *(float4*)&As[nxt][arow[i] * LDA_S + ac4 * 4] = ta[i];
#pragma unroll
      for (int i = 0; i < 4; ++i) {
        Bs[nxt][(bc4 * 4 + 0) * LDB_S + brow[i]] = tb[i].x;
        Bs[nxt][(bc4 * 4 + 1) * LDB_S + brow[i]] = tb[i].y;
        Bs[nxt][(bc4 * 4 + 2) * LDB_S + brow[i]] = tb[i].z;
        Bs[nxt][(bc4 * 4 + 3) * LDB_S + brow[i]] = tb[i].w;
      }
      __syncthreads();
    }
    cur = nxt;
  }

  // D layout: lane l16 -> N col, VGPR r -> M = r + 8*hf
#pragma unroll
  for (int mi = 0; mi < 2; ++mi) {
#pragma unroll
    for (int ni = 0; ni < 4; ++ni) {
      const int gn = blockN + wn + ni * 16 + l16;
      const float bv = bias ? bias[gn] : 0.f;
      const int gm0 = blockM + wm + mi * 16 + hf * 8;
#pragma unroll
      for (int rr = 0; rr < 8; ++rr)
        C[(size_t)(gm0 + rr) * N + gn] = acc[mi][ni][rr] + bv;
    }
  }
}

// ------------------------------------------------------------------
// q logits: ql[bh,n] = scale * dot(q[b,h,n,:], w_q_logits)
// ------------------------------------------------------------------
__global__ __launch_bounds__(256) void q_logits_kernel(
    const float* __restrict__ qkv, const float* __restrict__ wql,
    float* __restrict__ out)
{
  const int bh = blockIdx.y, b = bh >> 3, h = bh & 7;
  __shared__ float coef[DHEAD];
  if (threadIdx.x < DHEAD) coef[threadIdx.x] = wql[threadIdx.x] * SCALE;
  __syncthreads();
  const int n = blockIdx.x * 256 + threadIdx.x;
  const size_t base = ((size_t)(b * NSEQ + n)) * QKV_W + h * DHEAD;
  float s = 0.f;
#pragma unroll
  for (int d = 0; d < DHEAD; ++d) s += qkv[base + d] * coef[d];
  out[(size_t)bh * NSEQ + n] = s;
}

// ------------------------------------------------------------------
// k logits: kl[bh,n] = scale * sum_d k[d] * gq[d] * w_k_logits[d>>1]
// ------------------------------------------------------------------
__global__ __launch_bounds__(256) void k_logits_kernel(
    const float* __restrict__ qkv, const float* __restrict__ gq,
    const float* __restrict__ wkl, float* __restrict__ out)
{
  const int bh = blockIdx.y, b = bh >> 3, h = bh & 7;
  __shared__ float coef[DHEAD];
  if (threadIdx.x < DHEAD) {
    int d = threadIdx.x;
    coef[d] = gq[bh * DHEAD + d] * wkl[d >> 1] * SCALE;
  }
  __syncthreads();
  const int n = blockIdx.x * 256 + threadIdx.x;
  const size_t base = ((size_t)(b * NSEQ + n)) * QKV_W + 512 + h * DHEAD;
  float s = 0.f;
#pragma unroll
  for (int d = 0; d < DHEAD; ++d) s += qkv[base + d] * coef[d];
  out[(size_t)bh * NSEQ + n] = s;
}

// ------------------------------------------------------------------
// softmax over n + weighted pool of the ROTATED tensor:
//   g[b,h,d] = sum_n softmax(lg)[n] * rot(t)[b,h,n,d], t at qkv col `off`
// One block per (b,h); 256 threads = (d:64, chunk:4).
// ------------------------------------------------------------------
__global__ __launch_bounds__(256) void softmax_pool_rotary(
    const float* __restrict__ qkv, const float* __restrict__ logits,
    float* __restrict__ gout, int off)
{
  const int bh = blockIdx.x, b = bh >> 3, h = bh & 7;
  const float* lg = logits + (size_t)bh * NSEQ;
  const int tid = threadIdx.x;

  __shared__ float red[256];
  float m = -3.4e38f;
  for (int n = tid; n < NSEQ; n += 256) m = fmaxf(m, lg[n]);
  red[tid] = m;
  __syncthreads();
  for (int s = 128; s > 0; s >>= 1) {
    if (tid < s) red[tid] = fmaxf(red[tid], red[tid + s]);
    __syncthreads();
  }
  const float maxv = red[0];

  const int d = tid & 63;
  const int c = tid >> 6;   // 0..3
  // rotary: freq index j = d>>1, inv_freq = 10000^(-(2j)/32); only d<32 rotated
  const float invf = (d < 32) ? powf(10000.f, -(float)(d & ~1) / 32.f) : 0.f;

  float acc = 0.f, wsum = 0.f;
  for (int n = c; n < NSEQ; n += 4) {
    const float w = __expf(lg[n] - maxv);
    const size_t base = ((size_t)(b * NSEQ + n)) * QKV_W + off + h * DHEAD;
    float val;
    if (d < 32) {
      const float ang = invf * (float)n;
      const float sn = __sinf(ang), cs = __cosf(ang);
      const float xe = qkv[base + (d & ~1)];
      const float xo = qkv[base + (d | 1)];
      val = (d & 1) ? (xo * cs + xe * sn) : (xe * cs - xo * sn);
    } else {
      val = qkv[base + d];
    }
    acc += w * val;
    wsum += w;
  }

  __shared__ float accS[4][DHEAD];
  __shared__ float wS[4];
  accS[c][d] = acc;
  if (d == 0) wS[c] = wsum;
  __syncthreads();
  if (tid < DHEAD) {
    const float t = accS[0][tid] + accS[1][tid] + accS[2][tid] + accS[3][tid];
    const float wt = wS[0] + wS[1] + wS[2] + wS[3];
    gout[bh * DHEAD + tid] = t / wt;
  }
}

// ------------------------------------------------------------------
// u[j] = v[2j]*gk[2j] + v[2j+1]*gk[2j+1]; r = u @ w_r + b_r + q,
// written in (B, N, H*D) layout for the output GEMM.
// Block 256 = (d:64, n-sub:4); grid (NSEQ/4, BH).
// ------------------------------------------------------------------
__global__ __launch_bounds__(256) void build_r_kernel(
    const float* __restrict__ qkv, const float* __restrict__ gk,
    const float* __restrict__ w_r, const float* __restrict__ b_r,
    float* __restrict__ r)
{
  const int bh = blockIdx.y, b = bh >> 3, h = bh & 7;
  const int d = threadIdx.x & 63;
  const int y = threadIdx.x >> 6;          // 0..3
  const int n = blockIdx.x * 4 + y;

  __shared__ float wrS[32 * DHEAD];
  __shared__ float uS[4][32];
  __shared__ float gkS[DHEAD];
  for (int i = threadIdx.x; i < 32 * DHEAD; i += 256) wrS[i] = w_r[i];
  if (threadIdx.x < DHEAD) gkS[threadIdx.x] = gk[bh * DHEAD + threadIdx.x];
  __syncthreads();

  const size_t base = ((size_t)(b * NSEQ + n)) * QKV_W + h * DHEAD;
  if (d < 32) {
    const float v0 = qkv[base + 1024 + 2 * d];
    const float v1 = qkv[base + 1024 + 2 * d + 1];
    uS[y][d] = v0 * gkS[2 * d] + v1 * gkS[2 * d + 1];
  }
  __syncthreads();

  float accv = b_r[d] + qkv[base + d];     // + q (pre-rotary), per reference
#pragma unroll
  for (int j = 0; j < 32; ++j) accv += uS[y][j] * wrS[j * DHEAD + d];
  r[((size_t)(b * NSEQ + n)) * DIM_C + h * DHEAD + d] = accv;
}

// ------------------------------------------------------------------
extern "C" void kernel_launch(void* const* d_in, const int* in_sizes, int n_in,
                              void* d_out, int out_size, void* d_ws, size_t ws_size,
                              hipStream_t stream) {
  const float* x     = (const float*)d_in[0];
  const float* w_qkv = (const float*)d_in[1];
  const float* wql   = (const float*)d_in[2];
  const float* wkl   = (const float*)d_in[3];
  const float* w_r   = (const float*)d_in[4];
  const float* b_r   = (const float*)d_in[5];
  const float* w_out = (const float*)d_in[6];
  const float* b_out = (const float*)d_in[7];
  float* out = (float*)d_out;

  const size_t M = (size_t)BATCH * NSEQ;              // 32768
  float* ws  = (float*)d_ws;
  float* qkv = ws;                                    // M*1536
  float* ql  = qkv + M * QKV_W;                       // BH*N
  float* kl  = ql + (size_t)BH * NSEQ;                // BH*N
  float* gq  = kl + (size_t)BH * NSEQ;                // BH*64
  float* gk  = gq + (size_t)BH * DHEAD;               // BH*64
  float* r   = gk + (size_t)BH * DHEAD;               // M*512

  // 1) qkv = x @ w_qkv     (M=32768, N=1536, K=512)
  gemm_wmma_f32<<<dim3(QKV_W / BN, M / BM), 256, 0, stream>>>(
      x, w_qkv, nullptr, qkv, (int)M, QKV_W, DIM_C);

  // 2) q logits
  q_logits_kernel<<<dim3(NSEQ / 256, BH), 256, 0, stream>>>(qkv, wql, ql);

  // 3) softmax(q logits) pooled over rotated q -> global_q
  softmax_pool_rotary<<<BH, 256, 0, stream>>>(qkv, ql, gq, 0);

  // 4) k logits (uses global_q)
  k_logits_kernel<<<dim3(NSEQ / 256, BH), 256, 0, stream>>>(qkv, gq, wkl, kl);

  // 5) softmax(k logits) pooled over rotated k -> global_k
  softmax_pool_rotary<<<BH, 256, 0, stream>>>(qkv, kl, gk, 512);

  // 6) r = (v .* global_k pair-sum) @ w_r + b_r + q, laid out (B,N,512)
  build_r_kernel<<<dim3(NSEQ / 4, BH), 256, 0, stream>>>(qkv, gk, w_r, b_r, r);

  // 7) out = r @ w_out + b_out   (M=32768, N=512, K=512)
  gemm_wmma_f32<<<dim3(DIM_C / BN, M / BM), 256, 0, stream>>>(
      r, w_out, b_out, out, (int)M, DIM_C, DIM_C);
}